// LoRALayer_51994874086014
// MI455X (gfx1250) — compile-verified
//
#include <hip/hip_runtime.h>

// LoRA fused kernel for MI455X (gfx1250, wave32).
// out = 2.0 * (x @ B) @ A  computed in two WMMA phases, never materializing B@A.
// x:[16384,4096] f32, B:[4096,8] f32, A:[8,4096] f32, out:[16384,4096] f32.
//
// Roofline: factored path is ~1 GFLOP fp32 vs 550 GFLOP for the reference's
// x @ (B@A); traffic-bound at ~512 MB (read x + write out) -> ~22 us at
// 23.3 TB/s. fp32 WMMA (V_WMMA_F32_16X16X4_F32) keeps reference numerics
// while moving the FMAs off the VALU issue path.

typedef __attribute__((ext_vector_type(2))) float v2f;
typedef __attribute__((ext_vector_type(8))) float v8f;

#define IN_F    4096
#define OUT_F   4096
#define RANK    8
#define SCALING 2.0f

// Block = 128 threads = 4 wave32; each block owns one 16-row tile of x/out.
// Phase 1: wave w computes partial T(16x8) over K-quarter w using
//          V_WMMA_F32_16X16X4_F32:  A-op = x tile (16x4), B-op = B tile (4x16,
//          cols 8..15 zero via multiplicative mask). 256 WMMAs per wave.
// Phase 2: wave w produces output-column quarter w:
//          out(16x16 chunk) = SCALING * (T(16x8) @ A(8x16 chunk)), 2 WMMAs/chunk.
__global__ __launch_bounds__(128) void lora_fused_wmma(
    const float* __restrict__ x, const float* __restrict__ A,
    const float* __restrict__ B, float* __restrict__ out)
{
    __shared__ float Tpart[4][16][RANK];   // per-wave K-partial of T, 2 KB

    const int tid  = threadIdx.x;
    const int lane = tid & 31;
    const int wave = tid >> 5;
    const int m    = lane & 15;   // row index (A-op) / col index (B-op, D)
    const int hi   = lane >> 4;   // selects K pair (operands) / row+8 (C,D)

    // ---------------- Phase 1: partial T = x @ B over this wave's K-quarter --
    const int rowBase = blockIdx.x * 16;
    const int kBase   = wave * (IN_F / 4);

    // A-operand layout (16x4 f32): lane {hi,m} holds {x[m][k+2hi], x[m][k+2hi+1]}
    const float* xp = x + (size_t)(rowBase + m) * IN_F + kBase + 2 * hi;
    // B-operand layout (4x16 f32): lane {hi,m} holds {B[k+2hi][m], B[k+2hi+1][m]}.
    // Clamp the column so every lane loads a valid address (no EXEC branching);
    // rank padding (cols 8..15 = 0) is applied with a multiplicative mask.
    const float  mask = (m < RANK) ? 1.0f : 0.0f;
    const float* bp = B + (size_t)(kBase + 2 * hi) * RANK + (m & (RANK - 1));

    v8f acc = {};
    #pragma unroll 8
    for (int k0 = 0; k0 < IN_F / 4; k0 += 4) {
        v2f xa = *(const v2f*)(xp + k0);
        v2f bb = { bp[k0 * RANK] * mask, bp[k0 * RANK + RANK] * mask };
        acc = __builtin_amdgcn_wmma_f32_16x16x4_f32(
                  false, xa, false, bb, (short)0, acc, false, false);
    }

    // D layout: acc[v] = T[v + 8*hi][m]; only cols m<8 are meaningful.
    if (m < RANK) {
        #pragma unroll
        for (int v = 0; v < 8; ++v)
            Tpart[wave][v + 8 * hi][m] = acc[v];
    }
    __syncthreads();

    // ---------------- Phase 2: out = SCALING * (T @ A), N-quarter per wave ---
    // A-operand for T(16x8): lane {hi,m} needs {T[m][k0+2hi], T[m][k0+2hi+1]}
    // for k0 = 0 (a0) and k0 = 4 (a1); sum the four K-partials.
    v2f a0 = {}, a1 = {};
    #pragma unroll
    for (int p = 0; p < 4; ++p) {
        a0 += *(const v2f*)&Tpart[p][m][2 * hi];
        a1 += *(const v2f*)&Tpart[p][m][4 + 2 * hi];
    }

    float* op = out + (size_t)rowBase * OUT_F;
    const int chunk0 = wave * (OUT_F / 16 / 4);

    for (int ch = 0; ch < OUT_F / 16 / 4; ++ch) {
        const int col = (chunk0 + ch) * 16 + m;
        // B-operand: lane {hi,m} holds {A[k0+2hi][col], A[k0+2hi+1][col]};
        // 16 consecutive lanes hit 64B-contiguous segments (L2-resident A).
        v2f b0 = { A[(size_t)(2 * hi)     * OUT_F + col],
                   A[(size_t)(2 * hi + 1) * OUT_F + col] };
        v2f b1 = { A[(size_t)(4 + 2 * hi) * OUT_F + col],
                   A[(size_t)(5 + 2 * hi) * OUT_F + col] };
        v8f c = {};
        c = __builtin_amdgcn_wmma_f32_16x16x4_f32(
                false, a0, false, b0, (short)0, c, false, false);
        c = __builtin_amdgcn_wmma_f32_16x16x4_f32(
                false, a1, false, b1, (short)0, c, false, false);
        // D layout: c[v] = out-tile[v + 8*hi][col]; stores coalesce into
        // 64B segments per register row.
        #pragma unroll
        for (int v = 0; v < 8; ++v)
            op[(size_t)(v + 8 * hi) * OUT_F + col] = SCALING * c[v];
    }
}

extern "C" void kernel_launch(void* const* d_in, const int* in_sizes, int n_in,
                              void* d_out, int out_size, void* d_ws, size_t ws_size,
                              hipStream_t stream) {
    const float* x = (const float*)d_in[0];   // [4,4096,4096] f32
    const float* A = (const float*)d_in[1];   // [8,4096] f32
    const float* B = (const float*)d_in[2];   // [4096,8] f32
    float* out = (float*)d_out;               // [4,4096,4096] f32

    const int mtotal = in_sizes[0] / IN_F;    // 16384 rows
    lora_fused_wmma<<<dim3(mtotal / 16), dim3(128), 0, stream>>>(x, A, B, out);
}